// PoseNetV3_30511447671640
// MI455X (gfx1250) — compile-verified
//
#include <hip/hip_runtime.h>

#define SUPPORT 7
#define KNN_MAX 11

typedef float v2f __attribute__((ext_vector_type(2)));
typedef float v8f __attribute__((ext_vector_type(8)));

// ---------------------------------------------------------------------------
// Normalize direction matrices (3, D) along axis 0 (norm0 in the reference).
// ---------------------------------------------------------------------------
__global__ void norm_dirs_kernel(const float* __restrict__ d,
                                 float* __restrict__ out, int D) {
  int c = blockIdx.x * blockDim.x + threadIdx.x;
  if (c >= D) return;
  float a = d[c], b = d[D + c], g = d[2 * D + c];
  float n = fmaxf(sqrtf(a * a + b * b + g * g), 1e-12f);
  out[c] = a / n;
  out[D + c] = b / n;
  out[2 * D + c] = g / n;
}

// ---------------------------------------------------------------------------
// Brute-force kNN: keep k+1 smallest, drop nearest (self).
// ---------------------------------------------------------------------------
__global__ void knn_kernel(const float* __restrict__ v, int* __restrict__ idx,
                           int B, int n, int k) {
  int t = blockIdx.x * blockDim.x + threadIdx.x;
  if (t >= B * n) return;
  int b = t / n, i = t - b * n;
  const float* vb = v + (long long)b * n * 3;
  float xi = vb[3 * i], yi = vb[3 * i + 1], zi = vb[3 * i + 2];
  float bd[KNN_MAX];
  int bi[KNN_MAX];
  int cnt = k + 1;
  for (int q = 0; q < cnt; ++q) { bd[q] = 3.0e38f; bi[q] = 0; }
  for (int j = 0; j < n; ++j) {
    float dx = vb[3 * j] - xi;
    float dy = vb[3 * j + 1] - yi;
    float dz = vb[3 * j + 2] - zi;
    float d = dx * dx + dy * dy + dz * dz;
    if (d < bd[cnt - 1]) {
      int p = cnt - 1;
      while (p > 0 && bd[p - 1] > d) {
        bd[p] = bd[p - 1]; bi[p] = bi[p - 1]; --p;
      }
      bd[p] = d; bi[p] = j;
    }
  }
  int* row = idx + (long long)t * k;
  for (int q = 1; q < cnt; ++q) row[q - 1] = bi[q];
}

// ---------------------------------------------------------------------------
// conv_surface: theta = relu(dn . dir), max over k, sum over support, relu.
// One block (64 threads == kernel_num) per point; dn staged in LDS.
// ---------------------------------------------------------------------------
__global__ __launch_bounds__(64) void conv_surface_kernel(
    const float* __restrict__ v, const int* __restrict__ idx,
    const float* __restrict__ dirn, float* __restrict__ fm0, int n, int k) {
  __shared__ float dns[KNN_MAX][3];
  int b = blockIdx.y, i = blockIdx.x, e = threadIdx.x;
  const float* vb = v + (long long)b * n * 3;
  if (e < k) {
    int j = idx[((long long)b * n + i) * k + e];
    float dx = vb[3 * j] - vb[3 * i];
    float dy = vb[3 * j + 1] - vb[3 * i + 1];
    float dz = vb[3 * j + 2] - vb[3 * i + 2];
    float nn = fmaxf(sqrtf(dx * dx + dy * dy + dz * dz), 1e-12f);
    dns[e][0] = dx / nn; dns[e][1] = dy / nn; dns[e][2] = dz / nn;
  }
  __syncthreads();
  const int D = SUPPORT * 64;
  float sum = 0.f;
  for (int s = 0; s < SUPPORT; ++s) {
    int col = s * 64 + e;
    float d0 = dirn[col], d1 = dirn[D + col], d2 = dirn[2 * D + col];
    float mx = 0.f;  // theta >= 0 after relu
    for (int kk = 0; kk < k; ++kk) {
      float th = fmaxf(dns[kk][0] * d0 + dns[kk][1] * d1 + dns[kk][2] * d2, 0.f);
      mx = fmaxf(mx, th);
    }
    sum += mx;
  }
  fm0[((long long)b * n + i) * 128 + e] = fmaxf(sum, 0.f);
}

// ---------------------------------------------------------------------------
// rgb branch: relu(rgb_w @ rgb_f + rgb_b) then BN, into fm0[..., 64:128].
// ---------------------------------------------------------------------------
__global__ void rgb_kernel(const float* __restrict__ rgb_f,
                           const float* __restrict__ w,
                           const float* __restrict__ bias,
                           const float* __restrict__ g,
                           const float* __restrict__ bb,
                           const float* __restrict__ m,
                           const float* __restrict__ var,
                           float* __restrict__ fm0, int n) {
  long long t = (long long)blockIdx.x * blockDim.x + threadIdx.x;
  if (t >= (long long)4 * n * 64) return;
  int o = (int)(t % 64);
  long long bn_ = t / 64;
  int i = (int)(bn_ % n);
  int b = (int)(bn_ / n);
  const float* f = rgb_f + (long long)b * 32 * n + i;
  float s = bias[o];
  for (int c = 0; c < 32; ++c) s += w[o * 32 + c] * f[(long long)c * n];
  s = fmaxf(s, 0.f);
  s = (s - m[o]) * rsqrtf(var[o] + 1e-5f) * g[o] + bb[o];
  fm0[((long long)b * n + i) * 128 + 64 + o] = s;
}

// ---------------------------------------------------------------------------
// F32 WMMA GEMM: C[M,N] = A[M,K] @ W[K,N] + bias[N].
// One wave32 owns a 16x64 strip of C: 4 accumulators share one A fragment,
// so each K-step issues 1 A load (b64) + 8 B loads feeding 4 back-to-back
// V_WMMA_F32_16X16X4_F32 ops (4x register-level A reuse).
// A layout: lane<16 -> row=l, K {k,k+1}; lane>=16 -> row=l, K {k+2,k+3}.
// B layout: lane<16 -> K {k,k+1},  N=l; lane>=16 -> K {k+2,k+3}, N=l.
// C layout: vgpr r -> row m0+r (+8 for upper half), col n0+(lane&15).
// ---------------------------------------------------------------------------
__global__ __launch_bounds__(256) void wmma_gemm_kernel(
    const float* __restrict__ A, const float* __restrict__ W,
    const float* __restrict__ bias, float* __restrict__ C,
    int M, int K, int N) {
  const int lane = threadIdx.x & 31;
  const int wave = threadIdx.x >> 5;
  const int stripsN = N >> 6;  // 64-wide output strips
  const long long stripsTotal = (long long)(M >> 4) * stripsN;
  long long strip = (long long)blockIdx.x * 8 + wave;
  if (strip >= stripsTotal) return;  // wave-uniform: EXEC stays all-ones
  const int sm = (int)(strip / stripsN);
  const int sn = (int)(strip % stripsN);
  const int m0 = sm << 4, n0 = sn << 6;
  const int half = lane >> 4;
  const int l = lane & 15;

  const float* arow = A + (long long)(m0 + l) * K + half * 2;
  const float* bbase = W + (long long)(half * 2) * N + n0 + l;

  v8f acc[4];
#pragma unroll
  for (int t = 0; t < 4; ++t) acc[t] = (v8f){0.f, 0.f, 0.f, 0.f, 0.f, 0.f, 0.f, 0.f};

  for (int k = 0; k < K; k += 4) {
    if (k + 8 < K) {
      __builtin_prefetch(arow + k + 8, 0, 0);
      __builtin_prefetch(bbase + (long long)(k + 8) * N, 0, 0);
      __builtin_prefetch(bbase + (long long)(k + 9) * N, 0, 0);
    }
    v2f a = *(const v2f*)(arow + k);  // 8B-aligned: K%4==0, half*2+k even
#pragma unroll
    for (int t = 0; t < 4; ++t) {
      v2f bv;
      bv.x = bbase[(long long)k * N + t * 16];
      bv.y = bbase[(long long)(k + 1) * N + t * 16];
      acc[t] = __builtin_amdgcn_wmma_f32_16x16x4_f32(
          false, a, false, bv, (short)0, acc[t], false, false);
    }
  }

#pragma unroll
  for (int t = 0; t < 4; ++t) {
    const float bs = bias[n0 + t * 16 + l];
    float* crow = C + (long long)(m0 + half * 8) * N + n0 + t * 16 + l;
#pragma unroll
    for (int r = 0; r < 8; ++r) crow[(long long)r * N] = acc[t][r] + bs;
  }
}

// ---------------------------------------------------------------------------
// conv_layer tail: center + sum_s max_k( relu(dn.dir) * supp ), then BN+relu.
// One block (cout threads) per point; dn + neighbor ids staged in LDS.
// ---------------------------------------------------------------------------
__global__ __launch_bounds__(512) void conv_post_kernel(
    const float* __restrict__ v, const int* __restrict__ idx,
    const float* __restrict__ fout, const float* __restrict__ dirn,
    const float* __restrict__ g, const float* __restrict__ bb,
    const float* __restrict__ m, const float* __restrict__ var,
    float* __restrict__ out, int n, int k, int cout, int do_bnrelu) {
  __shared__ float dns[KNN_MAX][3];
  __shared__ int nbs[KNN_MAX];
  int b = blockIdx.y, i = blockIdx.x, e = threadIdx.x;
  const float* vb = v + (long long)b * n * 3;
  if (e < k) {
    int j = idx[((long long)b * n + i) * k + e];
    nbs[e] = j;
    float dx = vb[3 * j] - vb[3 * i];
    float dy = vb[3 * j + 1] - vb[3 * i + 1];
    float dz = vb[3 * j + 2] - vb[3 * i + 2];
    float nn = fmaxf(sqrtf(dx * dx + dy * dy + dz * dz), 1e-12f);
    dns[e][0] = dx / nn; dns[e][1] = dy / nn; dns[e][2] = dz / nn;
  }
  __syncthreads();
  const int D = SUPPORT * cout;
  const int fstride = 8 * cout;
  float acc = fout[((long long)b * n + i) * fstride + e];  // center
  for (int s = 0; s < SUPPORT; ++s) {
    int col = s * cout + e;
    float d0 = dirn[col], d1 = dirn[D + col], d2 = dirn[2 * D + col];
    float mx = -3.0e38f;
    for (int kk = 0; kk < k; ++kk) {
      float th = fmaxf(dns[kk][0] * d0 + dns[kk][1] * d1 + dns[kk][2] * d2, 0.f);
      float sp = fout[((long long)b * n + nbs[kk]) * fstride + cout + col];
      mx = fmaxf(mx, th * sp);
    }
    acc += mx;
  }
  if (do_bnrelu) {
    acc = (acc - m[e]) * rsqrtf(var[e] + 1e-5f) * g[e] + bb[e];
    acc = fmaxf(acc, 0.f);
  }
  out[((long long)b * n + i) * cout + e] = acc;
}

// ---------------------------------------------------------------------------
// pool: pf = max over 4 nbrs of f, subsample with deterministic permutation.
// n is a power of two, so (997*j + off) & (n-1) is a bijection.
// ---------------------------------------------------------------------------
__global__ __launch_bounds__(256) void pool_kernel(
    const float* __restrict__ v_in, const float* __restrict__ f_in,
    const int* __restrict__ idx4, float* __restrict__ v_out,
    float* __restrict__ f_out, int n, int m, int C, int seed) {
  int b = blockIdx.y, j = blockIdx.x, c = threadIdx.x;
  int src = (int)(((long long)997 * j + (long long)seed * 131) & (n - 1));
  const int* nb = idx4 + ((long long)b * n + src) * 4;
  float mx = -3.0e38f;
  for (int kk = 0; kk < 4; ++kk) {
    int q = nb[kk];
    mx = fmaxf(mx, f_in[((long long)b * n + q) * C + c]);
  }
  f_out[((long long)b * m + j) * C + c] = mx;
  if (c < 3) v_out[((long long)b * m + j) * 3 + c] = v_in[((long long)b * n + src) * 3 + c];
}

// ---------------------------------------------------------------------------
// nearest_idx: brute-force argmin over src per target.
// ---------------------------------------------------------------------------
__global__ void nearest_kernel(const float* __restrict__ tgt,
                               const float* __restrict__ src,
                               int* __restrict__ out, int B, int nt, int ns) {
  int t = blockIdx.x * blockDim.x + threadIdx.x;
  if (t >= B * nt) return;
  int b = t / nt, i = t - b * nt;
  const float* tb = tgt + ((long long)b * nt + i) * 3;
  const float* sb = src + (long long)b * ns * 3;
  float x = tb[0], y = tb[1], z = tb[2];
  float best = 3.0e38f;
  int bi = 0;
  for (int s = 0; s < ns; ++s) {
    float dx = sb[3 * s] - x, dy = sb[3 * s + 1] - y, dz = sb[3 * s + 2] - z;
    float d = dx * dx + dy * dy + dz * dz;
    if (d < best) { best = d; bi = s; }
  }
  out[t] = bi;
}

// ---------------------------------------------------------------------------
// f_global = max over points of fm4 (B, rows, 512).
// ---------------------------------------------------------------------------
__global__ void globalmax_kernel(const float* __restrict__ fm4,
                                 float* __restrict__ fg, int rows) {
  int t = blockIdx.x * blockDim.x + threadIdx.x;
  if (t >= 4 * 512) return;
  int b = t / 512, c = t - b * 512;
  const float* base = fm4 + (long long)b * rows * 512 + c;
  float mx = -3.0e38f;
  for (int r = 0; r < rows; ++r) mx = fmaxf(mx, base[(long long)r * 512]);
  fg[t] = mx;
}

// ---------------------------------------------------------------------------
// Final concat + transpose into feat (B,1280,n) and fuse (B,1792,n).
// ---------------------------------------------------------------------------
__global__ __launch_bounds__(256) void assemble_kernel(
    const float* __restrict__ fm0, const float* __restrict__ fm1,
    const float* __restrict__ fm2, const float* __restrict__ fm3,
    const float* __restrict__ fm4, const float* __restrict__ fglob,
    const int* __restrict__ ni1, const int* __restrict__ ni2,
    float* __restrict__ feat, float* __restrict__ fuse,
    int n, int n1, int n2) {
  int b = blockIdx.y, i = blockIdx.x;
  long long row = (long long)b * n + i;
  long long r1 = (long long)b * n1 + ni1[row];
  long long r2 = (long long)b * n2 + ni2[row];
  for (int c = threadIdx.x; c < 1792; c += blockDim.x) {
    float val;
    if (c < 128)        val = fm0[row * 128 + c];
    else if (c < 256)   val = fm1[row * 128 + (c - 128)];
    else if (c < 512)   val = fm2[r1 * 256 + (c - 256)];
    else if (c < 768)   val = fm3[r1 * 256 + (c - 512)];
    else if (c < 1280)  val = fm4[r2 * 512 + (c - 768)];
    else                val = fglob[(long long)b * 512 + (c - 1280)];
    if (c < 1280) feat[((long long)b * 1280 + c) * n + i] = val;
    fuse[((long long)b * 1792 + c) * n + i] = val;
  }
}

// ---------------------------------------------------------------------------
extern "C" void kernel_launch(void* const* d_in, const int* in_sizes, int n_in,
                              void* d_out, int out_size, void* d_ws, size_t ws_size,
                              hipStream_t stream) {
  (void)in_sizes; (void)n_in; (void)out_size; (void)ws_size;
  const int B = 4, N0 = 2048, N1 = 512, N2 = 128;

  const float* vertices = (const float*)d_in[0];
  const float* rgb_f    = (const float*)d_in[1];
  const float* dir0     = (const float*)d_in[2];
  const float* w1 = (const float*)d_in[3];
  const float* b1 = (const float*)d_in[4];
  const float* d1 = (const float*)d_in[5];
  const float* w2 = (const float*)d_in[6];
  const float* b2 = (const float*)d_in[7];
  const float* d2 = (const float*)d_in[8];
  const float* w3 = (const float*)d_in[9];
  const float* b3 = (const float*)d_in[10];
  const float* d3 = (const float*)d_in[11];
  const float* w4 = (const float*)d_in[12];
  const float* b4 = (const float*)d_in[13];
  const float* d4 = (const float*)d_in[14];
  const float* rgb_w = (const float*)d_in[15];
  const float* rgb_b = (const float*)d_in[16];
  const float* rbn_g = (const float*)d_in[17];
  const float* rbn_b = (const float*)d_in[18];
  const float* rbn_m = (const float*)d_in[19];
  const float* rbn_v = (const float*)d_in[20];
  const float* bn1_g = (const float*)d_in[21];
  const float* bn1_b = (const float*)d_in[22];
  const float* bn1_m = (const float*)d_in[23];
  const float* bn1_v = (const float*)d_in[24];
  const float* bn2_g = (const float*)d_in[25];
  const float* bn2_b = (const float*)d_in[26];
  const float* bn2_m = (const float*)d_in[27];
  const float* bn2_v = (const float*)d_in[28];
  const float* bn3_g = (const float*)d_in[29];
  const float* bn3_b = (const float*)d_in[30];
  const float* bn3_m = (const float*)d_in[31];
  const float* bn3_v = (const float*)d_in[32];

  float* feat = (float*)d_out;                         // (4,1280,2048)
  float* fuse = feat + (long long)B * 1280 * N0;       // (4,1792,2048)

  // ---- workspace carve-up (float slots; fout reused across the 4 GEMMs) ---
  float* p = (float*)d_ws;
  size_t off = 0;
  auto alloc = [&](size_t cnt) { float* r = p + off; off += cnt; return r; };
  float* dirn0 = alloc(3 * SUPPORT * 64);
  float* dirn1 = alloc(3 * SUPPORT * 128);
  float* dirn2 = alloc(3 * SUPPORT * 256);
  float* dirn3 = alloc(3 * SUPPORT * 256);
  float* dirn4 = alloc(3 * SUPPORT * 512);
  float* fm0  = alloc((size_t)B * N0 * 128);
  float* fout = alloc((size_t)B * N0 * 1024);  // >= all later fout sizes
  float* fm1  = alloc((size_t)B * N0 * 128);
  float* v1   = alloc((size_t)B * N1 * 3);
  float* fp1  = alloc((size_t)B * N1 * 128);
  float* fm2  = alloc((size_t)B * N1 * 256);
  float* fm3  = alloc((size_t)B * N1 * 256);
  float* v2   = alloc((size_t)B * N2 * 3);
  float* fp2  = alloc((size_t)B * N2 * 256);
  float* fm4  = alloc((size_t)B * N2 * 512);
  float* fglob = alloc((size_t)B * 512);
  int* idx0  = (int*)alloc((size_t)B * N0 * 10);
  int* idx4a = (int*)alloc((size_t)B * N0 * 4);
  int* idx1  = (int*)alloc((size_t)B * N1 * 10);
  int* idx4b = (int*)alloc((size_t)B * N1 * 4);
  int* idx2  = (int*)alloc((size_t)B * N2 * 10);
  int* ni1   = (int*)alloc((size_t)B * N0);
  int* ni2   = (int*)alloc((size_t)B * N0);

  auto gemm = [&](const float* A, const float* W, const float* bias, float* C,
                  int M, int K, int N) {
    long long strips = (long long)(M / 16) * (N / 64);
    int blocks = (int)((strips + 7) / 8);
    wmma_gemm_kernel<<<blocks, 256, 0, stream>>>(A, W, bias, C, M, K, N);
  };

  // ---- stage 0: normalized directions, knn, conv_surface, rgb ----
  norm_dirs_kernel<<<(SUPPORT * 64 + 63) / 64, 64, 0, stream>>>(dir0, dirn0, SUPPORT * 64);
  norm_dirs_kernel<<<(SUPPORT * 128 + 63) / 64, 64, 0, stream>>>(d1, dirn1, SUPPORT * 128);
  norm_dirs_kernel<<<(SUPPORT * 256 + 63) / 64, 64, 0, stream>>>(d2, dirn2, SUPPORT * 256);
  norm_dirs_kernel<<<(SUPPORT * 256 + 63) / 64, 64, 0, stream>>>(d3, dirn3, SUPPORT * 256);
  norm_dirs_kernel<<<(SUPPORT * 512 + 63) / 64, 64, 0, stream>>>(d4, dirn4, SUPPORT * 512);

  knn_kernel<<<(B * N0 + 127) / 128, 128, 0, stream>>>(vertices, idx0, B, N0, 10);
  conv_surface_kernel<<<dim3(N0, B), 64, 0, stream>>>(vertices, idx0, dirn0, fm0, N0, 10);
  rgb_kernel<<<((long long)B * N0 * 64 + 255) / 256, 256, 0, stream>>>(
      rgb_f, rgb_w, rgb_b, rbn_g, rbn_b, rbn_m, rbn_v, fm0, N0);

  // ---- layer 1 ----
  gemm(fm0, w1, b1, fout, B * N0, 128, (SUPPORT + 1) * 128);
  conv_post_kernel<<<dim3(N0, B), 128, 0, stream>>>(
      vertices, idx0, fout, dirn1, bn1_g, bn1_b, bn1_m, bn1_v, fm1, N0, 10, 128, 1);

  // ---- pool 1 ----
  knn_kernel<<<(B * N0 + 127) / 128, 128, 0, stream>>>(vertices, idx4a, B, N0, 4);
  pool_kernel<<<dim3(N1, B), 128, 0, stream>>>(vertices, fm1, idx4a, v1, fp1, N0, N1, 128, 1);
  knn_kernel<<<(B * N1 + 127) / 128, 128, 0, stream>>>(v1, idx1, B, N1, 10);

  // ---- layers 2, 3 ----
  gemm(fp1, w2, b2, fout, B * N1, 128, (SUPPORT + 1) * 256);
  conv_post_kernel<<<dim3(N1, B), 256, 0, stream>>>(
      v1, idx1, fout, dirn2, bn2_g, bn2_b, bn2_m, bn2_v, fm2, N1, 10, 256, 1);
  gemm(fm2, w3, b3, fout, B * N1, 256, (SUPPORT + 1) * 256);
  conv_post_kernel<<<dim3(N1, B), 256, 0, stream>>>(
      v1, idx1, fout, dirn3, bn3_g, bn3_b, bn3_m, bn3_v, fm3, N1, 10, 256, 1);

  // ---- pool 2 ----
  knn_kernel<<<(B * N1 + 127) / 128, 128, 0, stream>>>(v1, idx4b, B, N1, 4);
  pool_kernel<<<dim3(N2, B), 256, 0, stream>>>(v1, fm3, idx4b, v2, fp2, N1, N2, 256, 2);
  knn_kernel<<<(B * N2 + 127) / 128, 128, 0, stream>>>(v2, idx2, B, N2, 10);

  // ---- layer 4 (no bn/relu) ----
  gemm(fp2, w4, b4, fout, B * N2, 256, (SUPPORT + 1) * 512);
  conv_post_kernel<<<dim3(N2, B), 512, 0, stream>>>(
      v2, idx2, fout, dirn4, nullptr, nullptr, nullptr, nullptr, fm4, N2, 10, 512, 0);

  // ---- global max + upsample indices + final assembly ----
  globalmax_kernel<<<(B * 512 + 255) / 256, 256, 0, stream>>>(fm4, fglob, N2);
  nearest_kernel<<<(B * N0 + 255) / 256, 256, 0, stream>>>(vertices, v1, ni1, B, N0, N1);
  nearest_kernel<<<(B * N0 + 255) / 256, 256, 0, stream>>>(vertices, v2, ni2, B, N0, N2);
  assemble_kernel<<<dim3(N0, B), 256, 0, stream>>>(
      fm0, fm1, fm2, fm3, fm4, fglob, ni1, ni2, feat, fuse, N0, N1, N2);
}